// GRACE_61881888801196
// MI455X (gfx1250) — compile-verified
//
#include <hip/hip_runtime.h>
#include <stdint.h>

#define N_NODES 100000
#define N_EDGES 1600000
#define CH 128
#define LDW 132      // padded LDS row stride (floats); 132*4=528 bytes, 16B-aligned rows
#define RT 5         // row-tiles (of 16 rows) per block -> 80 rows/block
#define ROWS (RT*16) // 80
// 100000 / 80 = 1250 blocks exactly

typedef __attribute__((ext_vector_type(2))) float v2f;
typedef __attribute__((ext_vector_type(8))) float v8f;

// ---------------- utility kernels ----------------

__global__ void k_fill(float* __restrict__ p, float v, int n) {
    int i = blockIdx.x * 256 + threadIdx.x;
    if (i < n) p[i] = v;
}

__global__ void k_deg_scatter(const long long* __restrict__ dst,
                              float* __restrict__ deg, int n) {
    int i = blockIdx.x * 256 + threadIdx.x;
    if (i < n) atomicAdd(&deg[(int)dst[i]], 1.0f);
}

__global__ void k_rsqrt_inplace(float* __restrict__ p, int n) {
    int i = blockIdx.x * 256 + threadIdx.x;
    if (i < n) p[i] = rsqrtf(p[i]);
}

// ---------------- fp32 WMMA GEMM: H[M x 128] = X[M x 128] @ W[128 x 128] ----------------
// One block = 80 rows (5 x 16-row tiles). 8 waves; wave w owns columns [16w, 16w+16)
// and keeps 5 accumulators (one per row tile), so each B fragment read from LDS is
// reused by 5 consecutive V_WMMA_F32_16X16X4_F32 ops.
//
// A (16x4 f32) lane layout: lanes 0-15 hold M=lane, K={0,1}; lanes 16-31 hold M=lane-16, K={2,3}.
// B (4x16 f32) lane layout: lanes 0-15 hold N=lane, K={0,1}; lanes 16-31 hold N=lane-16, K={2,3}.
// C/D (16x16 f32): VGPR i -> M = i + (lane<16 ? 0 : 8), N = lane&15.

__global__ __launch_bounds__(256) void k_gemm(const float* __restrict__ X,
                                              const float* __restrict__ W,
                                              float* __restrict__ H) {
    __shared__ float Ws[128 * LDW];    // 67.6 KB
    __shared__ float Xs[ROWS * LDW];   // 42.2 KB

    const int t    = threadIdx.x;
    const int wave = t >> 5;
    const int lane = t & 31;
    const long rowBase = (long)blockIdx.x * ROWS;

    // Cooperative load of W (128x128) into padded LDS: 4096 float4's / 256 threads.
    #pragma unroll
    for (int i = 0; i < 16; ++i) {
        int j = t + i * 256;       // float4 slot 0..4095
        int r = j >> 5;            // row 0..127
        int q = j & 31;            // quad within row
        float4 v = ((const float4*)(W + (long)r * 128))[q];
        *(float4*)&Ws[r * LDW + q * 4] = v;
    }
    // Cooperative load of the 80x128 X slab: 2560 float4's / 256 threads.
    #pragma unroll
    for (int i = 0; i < 10; ++i) {
        int j = t + i * 256;       // 0..2559
        int r = j >> 5;            // row 0..79
        int q = j & 31;
        float4 v = ((const float4*)(X + (rowBase + r) * 128))[q];
        *(float4*)&Xs[r * LDW + q * 4] = v;
    }
    __syncthreads();

    const int mrow = lane & 15;
    const int kb   = (lane >> 4) * 2;          // 0 for lanes 0-15, 2 for lanes 16-31
    const int col  = (lane & 15) + wave * 16;  // output column this lane feeds (B frag)

    v8f acc[RT];
    #pragma unroll
    for (int tt = 0; tt < RT; ++tt)
        acc[tt] = (v8f){0.f, 0.f, 0.f, 0.f, 0.f, 0.f, 0.f, 0.f};

    #pragma unroll
    for (int k0 = 0; k0 < 128; k0 += 4) {
        v2f b;
        b.x = Ws[(k0 + kb) * LDW + col];
        b.y = Ws[(k0 + kb + 1) * LDW + col];
        #pragma unroll
        for (int tt = 0; tt < RT; ++tt) {
            v2f a;
            a.x = Xs[(tt * 16 + mrow) * LDW + k0 + kb];
            a.y = Xs[(tt * 16 + mrow) * LDW + k0 + kb + 1];
            // 8 args: (neg_a, A, neg_b, B, c_mod, C, reuse_a, reuse_b)
            acc[tt] = __builtin_amdgcn_wmma_f32_16x16x4_f32(
                false, a, false, b, (short)0, acc[tt], false, false);
        }
    }

    const int moff = (lane >> 4) * 8;   // lanes 16-31 hold rows M=8..15 of each tile
    const int ncol = (lane & 15) + wave * 16;
    #pragma unroll
    for (int tt = 0; tt < RT; ++tt) {
        #pragma unroll
        for (int i = 0; i < 8; ++i) {
            long m = rowBase + tt * 16 + i + moff;
            H[m * 128 + ncol] = acc[tt][i];
        }
    }
}

// ---------------- edge gather-scale-scatter (one wave per edge) ----------------
// Edge id is forced wave-uniform (readfirstlane) so edge indices and the two
// dis[] values lower to scalar loads instead of 32 redundant vector loads.

__global__ __launch_bounds__(256) void k_edge_scatter(
    const long long* __restrict__ src, const long long* __restrict__ dst,
    const float* __restrict__ dis, const float* __restrict__ h,
    float* __restrict__ y, int nEdges) {
    int wav = __builtin_amdgcn_readfirstlane(threadIdx.x >> 5);
    int e = blockIdx.x * 8 + wav;
    if (e >= nEdges) return;
    int s = (int)src[e];
    int d = (int)dst[e];
    float nrm = dis[s] * dis[d];
    int lane = threadIdx.x & 31;
    float4 v = ((const float4*)(h + (long)s * CH))[lane];  // 32 lanes x 4 ch = 128 ch
    float* o = y + (long)d * CH + lane * 4;
    atomicAdd(o + 0, v.x * nrm);
    atomicAdd(o + 1, v.y * nrm);
    atomicAdd(o + 2, v.z * nrm);
    atomicAdd(o + 3, v.w * nrm);
}

// ---------------- self-loop + bias + relu (fused) ----------------

__global__ void k_finalize(float* __restrict__ y, const float* __restrict__ h,
                           const float* __restrict__ dis, const float* __restrict__ b) {
    long i = (long)blockIdx.x * 256 + threadIdx.x;  // grid covers exactly N*CH
    int node = (int)(i >> 7);
    int c = (int)(i & 127);
    float dd = dis[node];
    float v = y[i] + h[i] * dd * dd + b[c];
    y[i] = v > 0.f ? v : 0.f;
}

// ---------------- launch ----------------

extern "C" void kernel_launch(void* const* d_in, const int* in_sizes, int n_in,
                              void* d_out, int out_size, void* d_ws, size_t ws_size,
                              hipStream_t stream) {
    (void)in_sizes; (void)n_in; (void)out_size; (void)ws_size;

    const float*     x  = (const float*)d_in[0];
    const long long* ei = (const long long*)d_in[1];   // int64 (2, E): row0=src, row1=dst
    const float*     W1 = (const float*)d_in[2];
    const float*     b1 = (const float*)d_in[3];
    const float*     W2 = (const float*)d_in[4];
    const float*     b2 = (const float*)d_in[5];
    const long long* srcI = ei;
    const long long* dstI = ei + N_EDGES;

    char*  ws  = (char*)d_ws;
    float* dis = (float*)ws;                             // 400 KB slot
    float* h   = (float*)(ws + (512 * 1024));            // 51.2 MB
    float* y   = h + (size_t)N_NODES * CH;               // 51.2 MB
    float* out = (float*)d_out;

    const int nodeBlocks = (N_NODES + 255) / 256;
    const int edgeBlocks = (N_EDGES + 255) / 256;
    const int edgeWaveBlocks = (N_EDGES + 7) / 8;
    const int featBlocks = (N_NODES * CH) / 256;         // 12.8M / 256 = exact
    const int gemmBlocks = N_NODES / ROWS;               // 1250 exact

    // deg -> dis = rsqrt(1 + indeg)
    k_fill<<<nodeBlocks, 256, 0, stream>>>(dis, 1.0f, N_NODES);
    k_deg_scatter<<<edgeBlocks, 256, 0, stream>>>(dstI, dis, N_EDGES);
    k_rsqrt_inplace<<<nodeBlocks, 256, 0, stream>>>(dis, N_NODES);

    // ---- layer 1 ----
    k_gemm<<<gemmBlocks, 256, 0, stream>>>(x, W1, h);
    k_fill<<<featBlocks, 256, 0, stream>>>(y, 0.0f, N_NODES * CH);
    k_edge_scatter<<<edgeWaveBlocks, 256, 0, stream>>>(srcI, dstI, dis, h, y, N_EDGES);
    k_finalize<<<featBlocks, 256, 0, stream>>>(y, h, dis, b1);

    // ---- layer 2 ----
    k_gemm<<<gemmBlocks, 256, 0, stream>>>(y, W2, h);
    k_fill<<<featBlocks, 256, 0, stream>>>(out, 0.0f, N_NODES * CH);
    k_edge_scatter<<<edgeWaveBlocks, 256, 0, stream>>>(srcI, dstI, dis, h, out, N_EDGES);
    k_finalize<<<featBlocks, 256, 0, stream>>>(out, h, dis, b2);
}